// ModelNew_5231270166622
// MI455X (gfx1250) — compile-verified
//
#include <hip/hip_runtime.h>

#define N 512
#define BM 32
#define BN 32
#define BK 32
#define LDA 34   // padded LDS row stride for A tile (floats): conflict-free frag reads
#define LDB 40   // padded LDS row stride for B tile (floats): half-waves hit disjoint banks

typedef float v2f __attribute__((ext_vector_type(2)));
typedef float v8f __attribute__((ext_vector_type(8)));

__device__ __forceinline__ int rowbase(int i) {
    // packed index of (i,i) in row-major packed upper-triangular storage
    return (i * (2 * N - i + 1)) >> 1;
}

__launch_bounds__(128, 4)
__global__ void triu_matmul_wmma_f32(const float* __restrict__ Ap,
                                     const float* __restrict__ Bp,
                                     float* __restrict__ C) {
    const int bi = blockIdx.y;          // 32-row block of C
    const int bj = blockIdx.x;          // 32-col block of C
    const int tid = threadIdx.x;        // 0..127, 4 waves of 32
    const int i0 = bi * BM;
    const int j0 = bj * BN;

    // Strictly-below-diagonal block: structurally zero output.
    // Each thread writes 8 *contiguous* floats -> merges into wide stores.
    if (bj < bi) {
        const int r = tid >> 2;                 // 0..31
        const int c0 = (tid & 3) * 8;           // 0,8,16,24
        float* dst = &C[(i0 + r) * N + (j0 + c0)];
        #pragma unroll
        for (int q = 0; q < 8; ++q) dst[q] = 0.0f;
        return;
    }

    __shared__ float As[BM * LDA];
    __shared__ float Bs[BK * LDB];

    const int lane = tid & 31;
    const int w    = tid >> 5;          // wave 0..3
    const int half = lane >> 4;         // 0: lanes 0-15, 1: lanes 16-31
    const int l    = lane & 15;
    const int m0   = (w >> 1) * 16;     // wave's 16x16 tile origin in block
    const int n0   = (w & 1) * 16;

    v8f acc = {};

    // Triangular support: A[i,k] needs k>=i (k >= i0), B[k,j] needs k<=j (k <= j0+31)
    const int kb_lo = bi;
    const int kb_hi = bj;

    // Per-thread staging coordinates (invariant across k-blocks)
    const int am = tid >> 2;                    // A: row within tile when t strides k
    // We stage with e = tid + t*128: m = e>>5, k = e&31  (8 elements per thread)

    for (int kb = kb_lo; kb <= kb_hi; ++kb) {
        const int k0 = kb * BK;

        // ---- Branchless unpack of A tile (rows i0.., cols k0..): ----
        // Clamp packed offset to the (always-valid) diagonal element, then
        // select zero after the load -> unconditional global_load + v_cndmask.
        float av[8];
        bool  amsk[8];
        #pragma unroll
        for (int t = 0; t < 8; ++t) {
            int e = tid + t * 128;              // 0..1023
            int m = e >> 5, k = e & 31;
            int i = i0 + m;
            int off = (k0 + k) - i;             // >=0 means on/above diagonal
            int offc = off >= 0 ? off : 0;      // clamped: idx always in-bounds
            amsk[t] = off >= 0;
            av[t] = Ap[rowbase(i) + offc];
        }
        float bv[8];
        bool  bmsk[8];
        #pragma unroll
        for (int t = 0; t < 8; ++t) {
            int e = tid + t * 128;
            int k = e >> 5, n = e & 31;
            int kk = k0 + k;
            int off = (j0 + n) - kk;            // >=0 means on/above diagonal
            int offc = off >= 0 ? off : 0;
            bmsk[t] = off >= 0;
            bv[t] = Bp[rowbase(kk) + offc];
        }
        #pragma unroll
        for (int t = 0; t < 8; ++t) {
            int e = tid + t * 128;
            int m = e >> 5, k = e & 31;
            As[m * LDA + k] = amsk[t] ? av[t] : 0.0f;
        }
        #pragma unroll
        for (int t = 0; t < 8; ++t) {
            int e = tid + t * 128;
            int k = e >> 5, n = e & 31;
            Bs[k * LDB + n] = bmsk[t] ? bv[t] : 0.0f;
        }
        __syncthreads();

        // 8 chained exact-f32 WMMA steps (K=4 each) over this 32-wide k block.
        // A frag: lanes 0-15 hold K={0,1}, lanes 16-31 hold K={2,3}, M = l.
        // B frag: lanes 0-15 hold K={0,1}, lanes 16-31 hold K={2,3}, N = l.
        #pragma unroll
        for (int kk = 0; kk < BK; kk += 4) {
            v2f a, b;
            const int ks = kk + 2 * half;
            a.x = As[(m0 + l) * LDA + ks + 0];
            a.y = As[(m0 + l) * LDA + ks + 1];
            b.x = Bs[(ks + 0) * LDB + n0 + l];
            b.y = Bs[(ks + 1) * LDB + n0 + l];
            acc = __builtin_amdgcn_wmma_f32_16x16x4_f32(
                /*neg_a=*/false, a, /*neg_b=*/false, b,
                /*c_mod=*/(short)0, acc, /*reuse_a=*/false, /*reuse_b=*/false);
        }
        __syncthreads();
    }

    (void)am;

    // C/D layout: lanes 0-15 -> M = r (VGPR r), lanes 16-31 -> M = r + 8; N = l.
    const int row0 = i0 + m0 + half * 8;
    const int col  = j0 + n0 + l;
    #pragma unroll
    for (int r = 0; r < 8; ++r) {
        C[(row0 + r) * N + col] = acc[r];
    }
}

extern "C" void kernel_launch(void* const* d_in, const int* in_sizes, int n_in,
                              void* d_out, int out_size, void* d_ws, size_t ws_size,
                              hipStream_t stream) {
    (void)in_sizes; (void)n_in; (void)out_size; (void)d_ws; (void)ws_size;
    const float* Ap = (const float*)d_in[0];
    const float* Bp = (const float*)d_in[1];
    float* C = (float*)d_out;
    dim3 grid(N / BN, N / BM);   // 16 x 16 blocks
    triu_matmul_wmma_f32<<<grid, 128, 0, stream>>>(Ap, Bp, C);
}